// HierarchyLoss_9801115369890
// MI455X (gfx1250) — compile-verified
//
#include <hip/hip_runtime.h>
#include <hip/hip_bf16.h>

typedef __attribute__((ext_vector_type(2))) float v2f;
typedef __attribute__((ext_vector_type(8))) float v8f;

#define HIER_WEIGHT 0.1f
#define NEG_BIG (-1.0e30f)

// ---------------------------------------------------------------------------
// Zero the scalar accumulator (d_out is poisoned, not re-zeroed by harness).
// ---------------------------------------------------------------------------
__global__ void hl_zero_kernel(float* out) {
    if (threadIdx.x == 0 && blockIdx.x == 0) out[0] = 0.0f;
}

// ---------------------------------------------------------------------------
// Cross-entropy: one block per batch row, single-pass online softmax
// (running max m, running sum s), wave32 shuffle reduction + LDS combine.
// Bandwidth-bound: streams the row once with float4 loads.
// ---------------------------------------------------------------------------
__global__ __launch_bounds__(512)
void hl_ce_kernel(const float* __restrict__ logits,
                  const int*   __restrict__ labels,
                  float*       __restrict__ out,
                  int N, float invB) {
    const int b   = blockIdx.x;
    const int tid = threadIdx.x;
    const int nthreads = blockDim.x;
    const float* __restrict__ row = logits + (size_t)b * (size_t)N;

    float m = NEG_BIG;
    float s = 0.0f;

    const int N4 = N >> 2;
    const float4* __restrict__ row4 = (const float4*)row;
    for (int i = tid; i < N4; i += nthreads) {
        float4 x = row4[i];
        float xm = fmaxf(fmaxf(x.x, x.y), fmaxf(x.z, x.w));
        float nm = fmaxf(m, xm);
        s = s * __expf(m - nm)
          + __expf(x.x - nm) + __expf(x.y - nm)
          + __expf(x.z - nm) + __expf(x.w - nm);
        m = nm;
    }
    for (int i = (N4 << 2) + tid; i < N; i += nthreads) {   // generic tail
        float x  = row[i];
        float nm = fmaxf(m, x);
        s = s * __expf(m - nm) + __expf(x - nm);
        m = nm;
    }

    // wave32 reduction of (m, s) pairs
    for (int off = 16; off > 0; off >>= 1) {
        float mo = __shfl_xor(m, off, 32);
        float so = __shfl_xor(s, off, 32);
        float nm = fmaxf(m, mo);
        s = s * __expf(m - nm) + so * __expf(mo - nm);
        m = nm;
    }

    __shared__ float sm[16];
    __shared__ float ss[16];
    const int wave = tid >> 5;
    const int lane = tid & 31;
    if (lane == 0) { sm[wave] = m; ss[wave] = s; }
    __syncthreads();

    if (tid == 0) {
        const int nwaves = nthreads >> 5;
        float M = sm[0], S = ss[0];
        for (int w = 1; w < nwaves; ++w) {
            float nm = fmaxf(M, sm[w]);
            S = S * __expf(M - nm) + ss[w] * __expf(sm[w] - nm);
            M = nm;
        }
        const int lbl = labels[b];
        const float x = row[lbl];
        const float ce = -(x - M - __logf(S));
        atomicAdd(out, ce * invB);
    }
}

// ---------------------------------------------------------------------------
// Hierarchy penalty: one edge per wave per iteration. Each lane reads 8
// consecutive floats of each gathered row (32 lanes x 8 = 256 = D, fully
// coalesced 1KB row reads), accumulates squared diffs into two fp32 partials.
// Final wave reduction done with V_WMMA_F32_16X16X4_F32:
//   A(16x4) holds all 64 per-lane partials, B = ones(4x16), D row-sums
//   collapse them; 8 adds + one shfl_xor(16) produce the full wave sum.
// EXEC is all-1s at the WMMA (loop condition is wave-uniform).
// ---------------------------------------------------------------------------
__global__ __launch_bounds__(256)
void hl_edge_kernel(const float* __restrict__ emb,
                    const int*   __restrict__ parent,
                    const int*   __restrict__ child,
                    float*       __restrict__ out,
                    int E, int D) {
    const int lane   = threadIdx.x & 31;
    const int wave   = (int)((blockIdx.x * blockDim.x + threadIdx.x) >> 5);
    const int nwaves = (int)((gridDim.x * blockDim.x) >> 5);

    float accA = 0.0f;  // squares from first float4 of each 8-float chunk
    float accB = 0.0f;  // squares from second float4

    for (int e = wave; e < E; e += nwaves) {
        const int p = parent[e];
        const int c = child[e];

        // prefetch next edge's rows (global_prefetch_b8); wave-uniform branch
        if (e + nwaves < E) {
            const int pn = parent[e + nwaves];
            const int cn = child[e + nwaves];
            __builtin_prefetch(emb + (size_t)pn * (size_t)D + lane * 8, 0, 0);
            __builtin_prefetch(emb + (size_t)cn * (size_t)D + lane * 8, 0, 0);
        }

        const float* __restrict__ prow = emb + (size_t)p * (size_t)D;
        const float* __restrict__ crow = emb + (size_t)c * (size_t)D;
        for (int j = lane * 8; j < D; j += 256) {  // D=256 -> exactly one pass
            float4 p0 = ((const float4*)(prow + j))[0];
            float4 p1 = ((const float4*)(prow + j))[1];
            float4 c0 = ((const float4*)(crow + j))[0];
            float4 c1 = ((const float4*)(crow + j))[1];
            float dx;
            dx = p0.x - c0.x; accA = fmaf(dx, dx, accA);
            dx = p0.y - c0.y; accA = fmaf(dx, dx, accA);
            dx = p0.z - c0.z; accA = fmaf(dx, dx, accA);
            dx = p0.w - c0.w; accA = fmaf(dx, dx, accA);
            dx = p1.x - c1.x; accB = fmaf(dx, dx, accB);
            dx = p1.y - c1.y; accB = fmaf(dx, dx, accB);
            dx = p1.z - c1.z; accB = fmaf(dx, dx, accB);
            dx = p1.w - c1.w; accB = fmaf(dx, dx, accB);
        }
    }

    // WMMA reduction: D[m][j] = sum_k A[m][k] with B = ones.
    v2f a;     a[0] = accA; a[1] = accB;
    v2f bones; bones[0] = 1.0f; bones[1] = 1.0f;
    v8f cz = {};
    v8f d = __builtin_amdgcn_wmma_f32_16x16x4_f32(
        /*neg_a=*/false, a, /*neg_b=*/false, bones,
        /*c_mod=*/(short)0, cz, /*reuse_a=*/false, /*reuse_b=*/false);

    float ssum = d[0] + d[1] + d[2] + d[3] + d[4] + d[5] + d[6] + d[7];
    ssum += __shfl_xor(ssum, 16, 32);   // halves hold rowsums 0-7 / 8-15

    if (lane == 0) atomicAdd(out, HIER_WEIGHT * ssum);
}

// ---------------------------------------------------------------------------
extern "C" void kernel_launch(void* const* d_in, const int* in_sizes, int n_in,
                              void* d_out, int out_size, void* d_ws, size_t ws_size,
                              hipStream_t stream) {
    const float* logits = (const float*)d_in[0];
    const int*   labels = (const int*)  d_in[1];
    const float* emb    = (const float*)d_in[2];
    const int*   parent = (const int*)  d_in[3];
    const int*   child  = (const int*)  d_in[4];
    float* out = (float*)d_out;

    const int B = in_sizes[1];             // 128
    const int N = in_sizes[0] / B;         // 100000
    const int D = in_sizes[2] / N;         // 256
    const int E = in_sizes[3];             // 99999

    hl_zero_kernel<<<1, 32, 0, stream>>>(out);
    hl_ce_kernel<<<B, 512, 0, stream>>>(logits, labels, out, N, 1.0f / (float)B);
    hl_edge_kernel<<<512, 256, 0, stream>>>(emb, parent, child, out, E, D);
}